// FastAttention_48155173323323
// MI455X (gfx1250) — compile-verified
//
#include <hip/hip_runtime.h>

typedef __attribute__((ext_vector_type(2))) float v2f;
typedef __attribute__((ext_vector_type(4))) float v4f;
typedef __attribute__((ext_vector_type(8))) float v8f;
typedef __attribute__((ext_vector_type(4))) int   v4i;

#define AS1 __attribute__((address_space(1)))
#define AS3 __attribute__((address_space(3)))

#if __has_builtin(__builtin_amdgcn_global_load_async_to_lds_b128)
#define HAVE_ASYNC_LDS 1
#else
#define HAVE_ASYNC_LDS 0
#endif

static __device__ __forceinline__ void async_copy_b128(const float* g, float* l) {
#if HAVE_ASYNC_LDS
  __builtin_amdgcn_global_load_async_to_lds_b128((AS1 v4i*)g, (AS3 v4i*)l, 0, 0);
#else
  *(v4f*)l = *(const v4f*)g;
#endif
}

static __device__ __forceinline__ void wait_async_all() {
#if HAVE_ASYNC_LDS
#if __has_builtin(__builtin_amdgcn_s_wait_asynccnt)
  __builtin_amdgcn_s_wait_asynccnt(0);
#else
  asm volatile("s_wait_asynccnt 0x0" ::: "memory");
#endif
#endif
}

#define B_     2
#define H_     16
#define S_     4096
#define D_     128
#define BH_    (B_ * H_)
#define SPLIT_ 16
#define CHUNK_ (S_ / SPLIT_)   // 256 s-rows per phase-1 workgroup
#define PANEL_ 32              // s-rows staged in LDS per double-buffer panel
#define PSTRIDE_ 136           // LDS row stride: 2*136 % 64 == 16 -> half-waves on disjoint banks

// ---------------------------------------------------------------------------
// Phase 1: partial KV[d,e] = sum_{s in chunk} K[s,d] * V[s,e]
// Double-buffered async staging of 32-row K/V panels into LDS; each wave owns
// a 2x4 block of 16x16 tiles (2 A-frags + 4 B-frags per 8 WMMAs).
// ---------------------------------------------------------------------------
__global__ __launch_bounds__(256) void kv_partial_kernel(
    const float* __restrict__ kmat, const float* __restrict__ vmat,
    float* __restrict__ partial) {
  __shared__ float Ks[2][PANEL_][PSTRIDE_];
  __shared__ float Vs[2][PANEL_][PSTRIDE_];

  const int bh    = blockIdx.x / SPLIT_;
  const int chnk  = blockIdx.x % SPLIT_;
  const int tid   = threadIdx.x;
  const int wave  = tid >> 5;
  const int lane  = tid & 31;
  const int half  = lane >> 4;    // K-half of the wave (ISA A/B layout)
  const int l     = lane & 15;    // M (A) / N (B) index
  const int mpair = wave >> 1;    // 0..3 -> M-tiles {2*mpair, 2*mpair+1}
  const int nquad = wave & 1;     // 0..1 -> N-tiles {4*nquad .. 4*nquad+3}

  const float* Kp = kmat + (size_t)bh * S_ * D_;
  const float* Vp = vmat + (size_t)bh * S_ * D_;
  const int s_begin = chnk * CHUNK_;

  auto issue_panel = [&](int p, int b) {
    const int sp = s_begin + p * PANEL_;
#pragma unroll
    for (int i = 0; i < 8; ++i) {             // 2048 float4 / 256 threads
      const int idx = tid + i * 256;
      const int mat = idx >> 10;              // 0: K panel, 1: V panel
      const int r   = (idx >> 5) & 31;
      const int c4  = (idx & 31) * 4;
      const float* g = (mat ? Vp : Kp) + (size_t)(sp + r) * D_ + c4;
      float* lp      = mat ? &Vs[b][r][c4] : &Ks[b][r][c4];
      async_copy_b128(g, lp);
    }
  };

  v8f acc[2][4] = {};
  issue_panel(0, 0);

  for (int p = 0; p < CHUNK_ / PANEL_; ++p) {
    const int b = p & 1;
    wait_async_all();       // my async loads landed in LDS
    __syncthreads();        // everyone's did; panel p-1 reads all done
    if (p + 1 < CHUNK_ / PANEL_) issue_panel(p + 1, b ^ 1);

#pragma unroll
    for (int so = 0; so < PANEL_; so += 4) {
      const int r0 = so + 2 * half;           // lanes 0-15: K=0,1  16-31: K=2,3
      v2f a[2];
#pragma unroll
      for (int mi = 0; mi < 2; ++mi) {
        const int dcol = mpair * 32 + mi * 16 + l;
        a[mi][0] = Ks[b][r0][dcol];
        a[mi][1] = Ks[b][r0 + 1][dcol];
      }
      v2f bb[4];
#pragma unroll
      for (int ni = 0; ni < 4; ++ni) {
        const int ecol = (nquad * 4 + ni) * 16 + l;
        bb[ni][0] = Vs[b][r0][ecol];
        bb[ni][1] = Vs[b][r0 + 1][ecol];
      }
#pragma unroll
      for (int mi = 0; mi < 2; ++mi)
#pragma unroll
        for (int ni = 0; ni < 4; ++ni)
          acc[mi][ni] = __builtin_amdgcn_wmma_f32_16x16x4_f32(
              false, a[mi], false, bb[ni], (short)0, acc[mi][ni], false, false);
    }
  }

  // C/D layout: VGPR r -> M = r (lanes 0-15), r+8 (lanes 16-31); N = l
  float* outp = partial + (size_t)(bh * SPLIT_ + chnk) * D_ * D_;
#pragma unroll
  for (int mi = 0; mi < 2; ++mi)
#pragma unroll
    for (int ni = 0; ni < 4; ++ni)
#pragma unroll
      for (int r = 0; r < 8; ++r) {
        const int drow = mpair * 32 + mi * 16 + r + 8 * half;
        const int ecol = (nquad * 4 + ni) * 16 + l;
        outp[(size_t)drow * D_ + ecol] = acc[mi][ni][r];
      }
}

// ---------------------------------------------------------------------------
// Phase 1b: deterministic reduce of the SPLIT_ partial KV buffers.
// ---------------------------------------------------------------------------
__global__ __launch_bounds__(256) void kv_reduce_kernel(
    const float* __restrict__ partial, float* __restrict__ kvf) {
  const int bh  = blockIdx.x >> 4;                 // 16 blocks per head
  const int blk = blockIdx.x & 15;
  const int i4  = blk * 256 + threadIdx.x;         // float4 index in [0, 4096)
  const size_t base = (size_t)bh * SPLIT_ * D_ * D_ + (size_t)i4 * 4;
  v4f s = {};
#pragma unroll
  for (int c = 0; c < SPLIT_; ++c)
    s += *(const v4f*)(partial + base + (size_t)c * D_ * D_);
  *(v4f*)(kvf + (size_t)bh * D_ * D_ + (size_t)i4 * 4) = s;
}

// ---------------------------------------------------------------------------
// Phase 2: out[s,e] = V[s,e] + sum_d Q[s,d] * KV[d,e]
// Q block (128x128) async-staged to LDS, row stride 132 (132 % 64 == 4) so
// column reads are bank-conflict-free; A-fragments are k-contiguous ->
// ds_load_b64. KV rows read direct (64 KB, hot in WGP$/L2). 2x4 blocking.
// ---------------------------------------------------------------------------
#define QSTRIDE_ 132

__global__ __launch_bounds__(256) void out_kernel(
    const float* __restrict__ qmat, const float* __restrict__ vmat,
    const float* __restrict__ kvf, float* __restrict__ outm) {
  __shared__ float Qs[128 * QSTRIDE_];

  const int bh    = blockIdx.x >> 5;   // 32 s-blocks per head
  const int sblk  = blockIdx.x & 31;
  const int sbase = sblk * 128;
  const int tid   = threadIdx.x;
  const int wave  = tid >> 5;
  const int lane  = tid & 31;
  const int half  = lane >> 4;
  const int l     = lane & 15;
  const int mpair = wave >> 1;         // 0..3
  const int nquad = wave & 1;          // 0..1

  const float* Qblk = qmat + ((size_t)bh * S_ + sbase) * D_;
#pragma unroll
  for (int i = 0; i < 16; ++i) {       // 4096 float4 / 256 threads
    const int idx = tid + i * 256;
    const int row = idx >> 5;
    const int c4  = (idx & 31) * 4;
    async_copy_b128(Qblk + (size_t)row * D_ + c4, &Qs[row * QSTRIDE_ + c4]);
  }
  wait_async_all();
  __syncthreads();

  const float* KVp = kvf + (size_t)bh * D_ * D_;
  v8f acc[2][4] = {};

  for (int k0 = 0; k0 < D_; k0 += 4) {
    const int kk = k0 + 2 * half;
    v2f a[2];
#pragma unroll
    for (int mi = 0; mi < 2; ++mi) {
      const int row = mpair * 32 + mi * 16 + l;
      a[mi] = *(const v2f*)&Qs[row * QSTRIDE_ + kk];   // ds_load_b64
    }
    v2f bb[4];
#pragma unroll
    for (int ni = 0; ni < 4; ++ni) {
      const int ecol = (nquad * 4 + ni) * 16 + l;
      bb[ni][0] = KVp[(size_t)(kk    ) * D_ + ecol];
      bb[ni][1] = KVp[(size_t)(kk + 1) * D_ + ecol];
    }
#pragma unroll
    for (int mi = 0; mi < 2; ++mi)
#pragma unroll
      for (int ni = 0; ni < 4; ++ni)
        acc[mi][ni] = __builtin_amdgcn_wmma_f32_16x16x4_f32(
            false, a[mi], false, bb[ni], (short)0, acc[mi][ni], false, false);
  }

  // Fused epilogue: out = V + acc
  const float* Vp = vmat + ((size_t)bh * S_ + sbase) * D_;
  float*       Op = outm + ((size_t)bh * S_ + sbase) * D_;
#pragma unroll
  for (int mi = 0; mi < 2; ++mi)
#pragma unroll
    for (int ni = 0; ni < 4; ++ni)
#pragma unroll
      for (int r = 0; r < 8; ++r) {
        const int row = mpair * 32 + mi * 16 + r + 8 * half;
        const int col = (nquad * 4 + ni) * 16 + l;
        Op[(size_t)row * D_ + col] = Vp[(size_t)row * D_ + col] + acc[mi][ni][r];
      }
}

extern "C" void kernel_launch(void* const* d_in, const int* in_sizes, int n_in,
                              void* d_out, int out_size, void* d_ws, size_t ws_size,
                              hipStream_t stream) {
  (void)in_sizes; (void)n_in; (void)out_size; (void)ws_size;
  const float* q = (const float*)d_in[0];
  const float* k = (const float*)d_in[1];
  const float* v = (const float*)d_in[2];
  float* out = (float*)d_out;

  // Workspace: [BH][SPLIT][D][D] partials (32 MiB) then [BH][D][D] final (2 MiB)
  float* partial = (float*)d_ws;
  float* kvf     = partial + (size_t)BH_ * SPLIT_ * D_ * D_;

  kv_partial_kernel<<<BH_ * SPLIT_, 256, 0, stream>>>(k, v, partial);
  kv_reduce_kernel <<<BH_ * 16,     256, 0, stream>>>(partial, kvf);
  out_kernel       <<<BH_ * 32,     256, 0, stream>>>(q, v, kvf, out);
}